// FraudDetectionQuantumModel_65481071405006
// MI455X (gfx1250) — compile-verified
//
#include <hip/hip_runtime.h>
#include <hip/hip_bf16.h>

// ---------------------------------------------------------------------------
// 12-qubit batched statevector sim, MI455X / gfx1250.
//   - one workgroup (256 thr = 8 wave32) per batch sample; full 32KB state in LDS
//   - encoding (all RY) built as real product state from 16-entry tables
//   - per-layer single-qubit gates fused per 4-wire group into 16x16 complex
//     unitaries, applied with V_WMMA_F32_16X16X4_F32 (4 k-chunks, complex = 4 wmma)
//   - 11-CNOT chain == Gray-code permutation  i -> i ^ (i>>1)
//   - Z expectations + linear head folded into one weighted |psi|^2 reduction
// ---------------------------------------------------------------------------

#define NW    12
#define NL    2
#define QDIM  4096
#define QBATCH 8192

typedef __attribute__((ext_vector_type(2))) float v2f;
typedef __attribute__((ext_vector_type(8))) float v8f;

static __device__ __forceinline__ v8f wmma4(v2f a, v2f b, v8f c) {
    // D(16x16,f32) += A(16x4,f32) * B(4x16,f32)
    return __builtin_amdgcn_wmma_f32_16x16x4_f32(false, a, false, b, (short)0, c,
                                                 false, false);
}

// --- build fused 16x16 complex gates: G[l][g] = kron_{k=0..3} (RY*RX)(theta) ---
__global__ void qgate_build(const float* __restrict__ params,
                            float* __restrict__ gr, float* __restrict__ gi) {
    int e = blockIdx.x * blockDim.x + threadIdx.x;
    if (e >= NL * 3 * 256) return;
    int l   = e / (3 * 256);
    int rem = e - l * (3 * 256);
    int g   = rem >> 8;
    int ij  = rem & 255;
    int i = ij >> 4, j = ij & 15;
    float re = 1.f, im = 0.f;
    for (int k = 0; k < 4; ++k) {
        int   w  = g * 4 + k;
        float th = params[l * NW + w] * 0.5f;
        float c  = cosf(th), s = sinf(th);
        int ib = (i >> (3 - k)) & 1;   // nibble bit 3 <-> wire 4g+0
        int jb = (j >> (3 - k)) & 1;
        // W = RY(t)*RX(t):  [[c^2+i s^2 , -cs(1+i)], [cs(1-i), c^2-i s^2]]
        float wr, wi;
        if (!ib && !jb)     { wr =  c * c; wi =  s * s; }
        else if (!ib && jb) { wr = -c * s; wi = -c * s; }
        else if (ib && !jb) { wr =  c * s; wi = -c * s; }
        else                { wr =  c * c; wi = -s * s; }
        float nr = re * wr - im * wi;
        float ni = re * wi + im * wr;
        re = nr; im = ni;
    }
    gr[e] = re;
    gi[e] = im;
}

// --- main simulator: one block per sample ---
__launch_bounds__(256)
__global__ void qsim(const float* __restrict__ x,
                     const float* __restrict__ head_w,
                     const float* __restrict__ head_b,
                     const float* __restrict__ gr,
                     const float* __restrict__ gi,
                     float* __restrict__ out) {
    __shared__ float Sr[QDIM];
    __shared__ float Si[QDIM];
    __shared__ float tabA[16], tabM[16], tabZ[16];   // encoding product tables
    __shared__ float cfA[16], cfM[16], cfZ[16];      // Z-expectation head weights
    __shared__ float trig[2 * NW];                   // cos, sin of x/2
    __shared__ float red[256];

    const int tid  = threadIdx.x;
    const int b    = blockIdx.x;
    const int lane = tid & 31;
    const int wave = tid >> 5;
    const int half = lane >> 4;   // 0: lanes 0-15, 1: lanes 16-31
    const int lm   = lane & 15;
    const int k0   = half * 2;    // K offset within a k-chunk per ISA A/B layout

    if (tid < NW) {
        float th = x[b * NW + tid] * 0.5f;
        trig[tid]      = cosf(th);
        trig[tid + NW] = sinf(th);
    }
    __syncthreads();

    if (tid < 48) {                       // encoding tables (product of 4 factors)
        int grp = tid >> 4, v = tid & 15;
        float p = 1.f;
        for (int k = 0; k < 4; ++k) {
            int w = grp * 4 + k;
            p *= ((v >> (3 - k)) & 1) ? trig[NW + w] : trig[w];
        }
        (grp == 0 ? tabA : grp == 1 ? tabM : tabZ)[v] = p;
    } else if (tid < 96) {                // head coefficient tables
        int grp = (tid - 48) >> 4, v = tid & 15;
        float s = 0.f;
        for (int k = 0; k < 4; ++k) {
            int w = grp * 4 + k;
            s += head_w[w] * (((v >> (3 - k)) & 1) ? -1.f : 1.f);
        }
        (grp == 0 ? cfA : grp == 1 ? cfM : cfZ)[v] = s;
    }
    __syncthreads();

    // --- encode: psi real product state, Si = 0 ---
    for (int k = 0; k < 16; ++k) {
        int i = tid * 16 + k;
        Sr[i] = tabA[(i >> 8) & 15] * tabM[(i >> 4) & 15] * tabZ[i & 15];
        Si[i] = 0.f;
    }
    __syncthreads();

    for (int l = 0; l < NL; ++l) {
        const float* Gr = gr + l * 3 * 256;
        const float* Gi = gi + l * 3 * 256;

        // ---- groups 0,1: D = U(16x16) x S_tile(16x16), 16 tiles ----
        for (int g = 0; g < 2; ++g) {
            const float* ur = Gr + g * 256;
            const float* ui = Gi + g * 256;
            v2f Ar[4], Av[4], An[4];           // U_re, U_im, -U_im in A layout
            for (int kk = 0; kk < 4; ++kk) {
                int kc = kk * 4 + k0;
                Ar[kk].x = ur[lm * 16 + kc];  Ar[kk].y = ur[lm * 16 + kc + 1];
                Av[kk].x = ui[lm * 16 + kc];  Av[kk].y = ui[lm * 16 + kc + 1];
                An[kk].x = -Av[kk].x;         An[kk].y = -Av[kk].y;
            }
            for (int tt = 0; tt < 2; ++tt) {
                int t = wave + tt * 8;
                v2f Br[4], Bi[4];              // state tile in B layout
                for (int kk = 0; kk < 4; ++kk) {
                    int kc = kk * 4 + k0;
                    int i0 = (g == 0) ? (kc * 256 + t * 16 + lm)
                                      : (t * 256 + kc * 16 + lm);
                    int i1 = (g == 0) ? ((kc + 1) * 256 + t * 16 + lm)
                                      : (t * 256 + (kc + 1) * 16 + lm);
                    Br[kk].x = Sr[i0];  Br[kk].y = Sr[i1];
                    Bi[kk].x = Si[i0];  Bi[kk].y = Si[i1];
                }
                v8f Cr = {}; v8f Ci = {};
                for (int kk = 0; kk < 4; ++kk) {
                    Cr = wmma4(Ar[kk], Br[kk], Cr);   // + Ur*Sr
                    Cr = wmma4(An[kk], Bi[kk], Cr);   // - Ui*Si
                    Ci = wmma4(Ar[kk], Bi[kk], Ci);   // + Ur*Si
                    Ci = wmma4(Av[kk], Br[kk], Ci);   // + Ui*Sr
                }
                for (int v = 0; v < 8; ++v) {         // C/D: M = v + 8*half, N = lm
                    int M = v + half * 8;
                    int o = (g == 0) ? (M * 256 + t * 16 + lm)
                                     : (t * 256 + M * 16 + lm);
                    Sr[o] = Cr[v];  Si[o] = Ci[v];
                }
            }
            __syncthreads();
        }

        // ---- group 2 (low nibble): D = S_tile(16x16) x U2^T ----
        {
            const float* ur = Gr + 2 * 256;
            const float* ui = Gi + 2 * 256;
            v2f Btr[4], Bti[4], Btn[4];        // U2^T (re, im, -im) in B layout
            for (int kk = 0; kk < 4; ++kk) {
                int kc = kk * 4 + k0;          // B[k,n] = U2[n,k]
                Btr[kk].x = ur[lm * 16 + kc];  Btr[kk].y = ur[lm * 16 + kc + 1];
                Bti[kk].x = ui[lm * 16 + kc];  Bti[kk].y = ui[lm * 16 + kc + 1];
                Btn[kk].x = -Bti[kk].x;        Btn[kk].y = -Bti[kk].y;
            }
            for (int tt = 0; tt < 2; ++tt) {
                int t = wave + tt * 8;
                v2f Asr[4], Asi[4];            // state tile rows in A layout
                int rbase = (t * 16 + lm) * 16;
                for (int kk = 0; kk < 4; ++kk) {
                    int kc = kk * 4 + k0;
                    Asr[kk].x = Sr[rbase + kc];  Asr[kk].y = Sr[rbase + kc + 1];
                    Asi[kk].x = Si[rbase + kc];  Asi[kk].y = Si[rbase + kc + 1];
                }
                v8f Cr = {}; v8f Ci = {};
                for (int kk = 0; kk < 4; ++kk) {
                    Cr = wmma4(Asr[kk], Btr[kk], Cr);
                    Cr = wmma4(Asi[kk], Btn[kk], Cr);
                    Ci = wmma4(Asr[kk], Bti[kk], Ci);
                    Ci = wmma4(Asi[kk], Btr[kk], Ci);
                }
                for (int v = 0; v < 8; ++v) {
                    int M = v + half * 8;
                    int o = (t * 16 + M) * 16 + lm;
                    Sr[o] = Cr[v];  Si[o] = Ci[v];
                }
            }
            __syncthreads();
        }

        // ---- CNOT chain == Gray permutation: new[i] = old[i ^ (i>>1)] ----
        {
            float vr[16], vi[16];
            for (int k = 0; k < 16; ++k) {
                int i = tid * 16 + k;
                int s = i ^ (i >> 1);
                vr[k] = Sr[s];  vi[k] = Si[s];
            }
            __syncthreads();
            for (int k = 0; k < 16; ++k) {
                int i = tid * 16 + k;
                Sr[i] = vr[k];  Si[i] = vi[k];
            }
            __syncthreads();
        }
    }

    // ---- weighted |psi|^2 reduction:  out = sum_i p_i * coef(i) + head_b ----
    float partial = 0.f;
    {
        float base = cfA[(tid >> 4) & 15] + cfM[tid & 15];
        for (int k = 0; k < 16; ++k) {
            int i = tid * 16 + k;
            float pr = Sr[i], pi = Si[i];
            partial += (pr * pr + pi * pi) * (base + cfZ[k]);
        }
    }
    red[tid] = partial;
    __syncthreads();
    for (int s = 128; s > 0; s >>= 1) {
        if (tid < s) red[tid] += red[tid + s];
        __syncthreads();
    }
    if (tid == 0) out[b] = red[0] + head_b[0];
}

extern "C" void kernel_launch(void* const* d_in, const int* in_sizes, int n_in,
                              void* d_out, int out_size, void* d_ws, size_t ws_size,
                              hipStream_t stream) {
    const float* x      = (const float*)d_in[0];   // [8192,12]
    const float* params = (const float*)d_in[1];   // [2,12]
    const float* head_w = (const float*)d_in[2];   // [1,12]
    const float* head_b = (const float*)d_in[3];   // [1]
    float* out = (float*)d_out;                    // [8192,1]

    float* gr = (float*)d_ws;                      // 1536 floats
    float* gi = gr + NL * 3 * 256;                 // 1536 floats (12 KB total)

    qgate_build<<<(NL * 3 * 256 + 255) / 256, 256, 0, stream>>>(params, gr, gi);
    qsim<<<QBATCH, 256, 0, stream>>>(x, head_w, head_b, gr, gi, out);
}